// SDF_15212774162966
// MI455X (gfx1250) — compile-verified
//
#include <hip/hip_runtime.h>
#include <math.h>

// P=1024 points, V=2048 verts, F=4096 faces (int64 indices).
// out[pt] = signed distance of the face minimizing |dist|, first index on ties.

#define BLOCK      256   // points per block (8 wave32)
#define FACE_TILE  128   // faces staged in LDS per block

#ifndef __has_builtin
#define __has_builtin(x) 0
#endif

// CDNA5 async global->LDS path (ASYNCcnt-tracked DMA into LDS).
#if __has_builtin(__builtin_amdgcn_global_load_async_to_lds_b64)
#define SDF_ASYNC_LDS 1
#else
#define SDF_ASYNC_LDS 0
#endif

#if SDF_ASYNC_LDS
typedef __attribute__((ext_vector_type(2))) int sdf_v2i;
typedef sdf_v2i __attribute__((address_space(1)))* sdf_gq_p;  // global qword ptr
typedef sdf_v2i __attribute__((address_space(3)))* sdf_lq_p;  // LDS qword ptr
#endif

__device__ __forceinline__ void stage_qword(const long long* g, long long* l) {
#if SDF_ASYNC_LDS
    __builtin_amdgcn_global_load_async_to_lds_b64(
        (sdf_gq_p)g, (sdf_lq_p)l, /*imm offset*/ 0, /*cpol*/ 0);
#else
    *l = *g;
#endif
}

__device__ __forceinline__ void wait_async_stage() {
#if SDF_ASYNC_LDS
#if __has_builtin(__builtin_amdgcn_s_wait_asynccnt)
    __builtin_amdgcn_s_wait_asynccnt(0);
#else
    asm volatile("s_wait_asynccnt 0x0" ::: "memory");
#endif
#endif
}

// ---------------------------------------------------------------- init packed
__global__ __launch_bounds__(256) void sdf_init_kernel(unsigned long long* __restrict__ packed,
                                                       int P) {
    int i = blockIdx.x * blockDim.x + threadIdx.x;
    if (i < P) packed[i] = ~0ull;
}

// ---------------------------------------------------------------- main kernel
__global__ __launch_bounds__(BLOCK) void sdf_main_kernel(
    const float* __restrict__ p,          // [P,3]
    const float* __restrict__ v,          // [V,3]
    const long long* __restrict__ f,      // [F,3] int64
    unsigned long long* __restrict__ packed, // [P] lexicographic (dist2, code)
    int P, int F)
{
    __shared__ long long fIdx[FACE_TILE * 3];  // raw index tile (3 KB), async-staged
    __shared__ float4 q0[FACE_TILE];  // p0.xyz , nn
    __shared__ float4 q1[FACE_TILE];  // p1.xyz , ae1
    __shared__ float4 q2[FACE_TILE];  // p2.xyz , ae2
    __shared__ float4 q3[FACE_TILE];  // n.xyz  , ae3
    __shared__ float4 q4[FACE_TILE];  // e1.xyz , 1/m1
    __shared__ float4 q5[FACE_TILE];  // e2.xyz , 1/m2
    __shared__ float4 q6[FACE_TILE];  // e3.xyz , 1/m3

    const int tid      = threadIdx.x;
    const int faceBase = blockIdx.y * FACE_TILE;
    const int nFaces   = (F - faceBase) < FACE_TILE ? (F - faceBase) : FACE_TILE;
    const int nQ       = 3 * nFaces;                 // qwords in the f tile
    const int pt       = blockIdx.x * BLOCK + tid;

    // ---- kick off async DMA of the face-index tile into LDS ----
    const long long* fTile = f + 3 * (size_t)faceBase;
    for (int e = tid; e < nQ; e += BLOCK)
        stage_qword(fTile + e, fIdx + e);

    // Overlap: fetch this thread's query point while the index DMA is in flight.
    float px = 0.0f, py = 0.0f, pz = 0.0f;
    if (pt < P) { px = p[3 * pt + 0]; py = p[3 * pt + 1]; pz = p[3 * pt + 2]; }

    wait_async_stage();       // this wave's ASYNCcnt -> 0 (its LDS writes landed)
    __syncthreads();          // all waves' staging visible

    // ---- per-face constants (one thread per face) ----
    if (tid < nFaces) {
        const long long i0 = fIdx[3 * tid + 0];
        const long long i1 = fIdx[3 * tid + 1];
        const long long i2 = fIdx[3 * tid + 2];
        const float p0x = v[3 * i0 + 0], p0y = v[3 * i0 + 1], p0z = v[3 * i0 + 2];
        const float p1x = v[3 * i1 + 0], p1y = v[3 * i1 + 1], p1z = v[3 * i1 + 2];
        const float p2x = v[3 * i2 + 0], p2y = v[3 * i2 + 1], p2z = v[3 * i2 + 2];

        const float e1x = p0x - p1x, e1y = p0y - p1y, e1z = p0z - p1z;
        const float e2x = p2x - p1x, e2y = p2y - p1y, e2z = p2z - p1z;
        const float e3x = p2x - p0x, e3y = p2y - p0y, e3z = p2z - p0z;

        // n = cross(p0-p1, p2-p1) = cross(e1, e2)
        const float nx = e1y * e2z - e1z * e2y;
        const float ny = e1z * e2x - e1x * e2z;
        const float nz = e1x * e2y - e1y * e2x;
        const float nn = nx * nx + ny * ny + nz * nz;

        const float ae1 = p0x * e1x + p0y * e1y + p0z * e1z;   // edge(a=p0,b=p1,e1)
        const float ae2 = p2x * e2x + p2y * e2y + p2z * e2z;   // edge(a=p2,b=p1,e2)
        const float ae3 = p2x * e3x + p2y * e3y + p2z * e3z;   // edge(a=p2,b=p0,e3)
        const float im1 = 1.0f / (e1x * e1x + e1y * e1y + e1z * e1z);
        const float im2 = 1.0f / (e2x * e2x + e2y * e2y + e2z * e2z);
        const float im3 = 1.0f / (e3x * e3x + e3y * e3y + e3z * e3z);

        q0[tid] = make_float4(p0x, p0y, p0z, nn);
        q1[tid] = make_float4(p1x, p1y, p1z, ae1);
        q2[tid] = make_float4(p2x, p2y, p2z, ae2);
        q3[tid] = make_float4(nx, ny, nz, ae3);
        q4[tid] = make_float4(e1x, e1y, e1z, im1);
        q5[tid] = make_float4(e2x, e2y, e2z, im2);
        q6[tid] = make_float4(e3x, e3y, e3z, im3);
    }
    __syncthreads();

    if (pt >= P) return;

    float        best2    = __builtin_huge_valf();   // squared distance
    unsigned int bestCode = 0xFFFFFFFFu;             // (faceIdx<<1)|signBit

    #pragma unroll 2
    for (int j = 0; j < nFaces; ++j) {
        const float4 a0 = q0[j], a1 = q1[j], a2 = q2[j], a3 = q3[j];
        const float4 a4 = q4[j], a5 = q5[j], a6 = q6[j];

        // tvec = p - p2 ; det = n . tvec ; dp^2 = det^2 * |n|^2
        const float tvx = px - a2.x, tvy = py - a2.y, tvz = pz - a2.z;
        const float det = a3.x * tvx + a3.y * tvy + a3.z * tvz;
        const float dp2 = det * det * a0.w;

        // pt_i = p_i - proj = (p_i - p) + n*det  (proj = p - n*det)
        const float ndx = a3.x * det, ndy = a3.y * det, ndz = a3.z * det;
        const float d0x = a0.x - px, d0y = a0.y - py, d0z = a0.z - pz;
        const float d1x = a1.x - px, d1y = a1.y - py, d1z = a1.z - pz;
        const float t0x = d0x + ndx, t0y = d0y + ndy, t0z = d0z + ndz;
        const float t1x = d1x + ndx, t1y = d1y + ndy, t1z = d1z + ndz;
        const float t2x = ndx - tvx, t2y = ndy - tvy, t2z = ndz - tvz;

        // u = cross(t1,t2); w = cross(t0,t1); vv = cross(t2,t0)
        const float ux = t1y * t2z - t1z * t2y;
        const float uy = t1z * t2x - t1x * t2z;
        const float uz = t1x * t2y - t1y * t2x;
        const float wx = t0y * t1z - t0z * t1y;
        const float wy = t0z * t1x - t0x * t1z;
        const float wz = t0x * t1y - t0y * t1x;
        const float vx = t2y * t0z - t2z * t0y;
        const float vy = t2z * t0x - t2x * t0z;
        const float vz = t2x * t0y - t2y * t0x;
        const float duv = ux * vx + uy * vy + uz * vz;
        const float duw = ux * wx + uy * wy + uz * wz;
        const bool inside = (duv >= 0.0f) && (duw >= 0.0f);

        // edge 1: c - p = (p0 - p) - s1*e1
        float s1 = (a1.w - (px * a4.x + py * a4.y + pz * a4.z)) * a4.w;
        s1 = fminf(fmaxf(s1, 0.0f), 1.0f);
        const float c1x = d0x - s1 * a4.x, c1y = d0y - s1 * a4.y, c1z = d0z - s1 * a4.z;
        const float dd1 = c1x * c1x + c1y * c1y + c1z * c1z;
        // edge 2: c - p = (p2 - p) - s2*e2 = -tvec - s2*e2
        float s2 = (a2.w - (px * a5.x + py * a5.y + pz * a5.z)) * a5.w;
        s2 = fminf(fmaxf(s2, 0.0f), 1.0f);
        const float c2x = -tvx - s2 * a5.x, c2y = -tvy - s2 * a5.y, c2z = -tvz - s2 * a5.z;
        const float dd2 = c2x * c2x + c2y * c2y + c2z * c2z;
        // edge 3: c - p = -tvec - s3*e3
        float s3 = (a3.w - (px * a6.x + py * a6.y + pz * a6.z)) * a6.w;
        s3 = fminf(fmaxf(s3, 0.0f), 1.0f);
        const float c3x = -tvx - s3 * a6.x, c3y = -tvy - s3 * a6.y, c3z = -tvz - s3 * a6.z;
        const float dd3 = c3x * c3x + c3y * c3y + c3z * c3z;

        const float dedge2 = fminf(dd1, fminf(dd2, dd3));
        const float r2     = inside ? dp2 : dedge2;

        const unsigned int code =
            ((unsigned int)(faceBase + j) << 1) | (det > 0.0f ? 1u : 0u);
        if (r2 < best2) { best2 = r2; bestCode = code; }  // strict '<' => first index wins
    }

    // Lexicographic (dist2, faceIdx) min across face-splits: IEEE bits of a
    // non-negative float are monotone as u32, face index sits in the low word.
    const unsigned long long pk =
        ((unsigned long long)__float_as_uint(best2) << 32) | (unsigned long long)bestCode;
    atomicMin(&packed[pt], pk);
}

// ---------------------------------------------------------------- finalize
__global__ __launch_bounds__(256) void sdf_final_kernel(
    const unsigned long long* __restrict__ packed, float* __restrict__ out, int P)
{
    int i = blockIdx.x * blockDim.x + threadIdx.x;
    if (i < P) {
        const unsigned long long pk = packed[i];
        const float b2   = __uint_as_float((unsigned int)(pk >> 32));
        const float sign = (pk & 1ull) ? 1.0f : -1.0f;
        out[i] = sqrtf(b2) * sign;
    }
}

// ---------------------------------------------------------------- launch
extern "C" void kernel_launch(void* const* d_in, const int* in_sizes, int n_in,
                              void* d_out, int out_size, void* d_ws, size_t ws_size,
                              hipStream_t stream) {
    const float*     p = (const float*)d_in[0];
    const float*     v = (const float*)d_in[1];
    const long long* f = (const long long*)d_in[2];
    float*           out = (float*)d_out;

    const int P = in_sizes[0] / 3;   // 1024
    const int F = in_sizes[2] / 3;   // 4096

    unsigned long long* packed = (unsigned long long*)d_ws;   // P * 8 bytes

    sdf_init_kernel<<<(P + 255) / 256, 256, 0, stream>>>(packed, P);

    dim3 grid((P + BLOCK - 1) / BLOCK, (F + FACE_TILE - 1) / FACE_TILE);
    sdf_main_kernel<<<grid, BLOCK, 0, stream>>>(p, v, f, packed, P, F);

    sdf_final_kernel<<<(P + 255) / 256, 256, 0, stream>>>(packed, out, P);
}